// DMDMachine_50740743634987
// MI455X (gfx1250) — compile-verified
//
#include <hip/hip_runtime.h>
#include <math.h>

// Problem constants
#define Bn 256
#define Pn 256
#define Tn 256
#define Ln 128
#define Hn 512

typedef float v8f __attribute__((ext_vector_type(8)));
typedef float v2f __attribute__((ext_vector_type(2)));

__device__ __forceinline__ v8f wmma_f32(v2f a, v2f b, v8f c) {
  // V_WMMA_F32_16X16X4_F32 : D = A(16x4) * B(4x16) + C(16x16), all f32
  return __builtin_amdgcn_wmma_f32_16x16x4_f32(false, a, false, b, (short)0, c,
                                               false, false);
}

// ---------------------------------------------------------------------------
// Encode: y[b,l,t] = (tanh(x[b,:,t]^T @ We1 + be1) @ We2 + be2)[l]
// grid (B, T/16), block 256 (8 waves). Each block: 16 t-rows.
// ---------------------------------------------------------------------------
__global__ __launch_bounds__(256) void k_encode(
    const float* __restrict__ x, const float* __restrict__ We1,
    const float* __restrict__ be1, const float* __restrict__ We2,
    const float* __restrict__ be2, float* __restrict__ y) {
  __shared__ float Xs[16 * Pn];   // 16KB : Xt tile (t-major rows, p cols)
  __shared__ float Hs[16 * Hn];   // 32KB : hidden tile
  const int b = blockIdx.x;
  const int t0 = blockIdx.y * 16;
  const int tid = threadIdx.x;
  for (int idx = tid; idx < 16 * Pn; idx += 256) {
    int m = idx & 15, p = idx >> 4;
    Xs[m * Pn + p] = x[((size_t)b * Pn + p) * Tn + t0 + m];
  }
  __syncthreads();
  const int wave = tid >> 5;
  const int lane = tid & 31;
  const int mrow = lane & 15;          // A-frag row
  const int koff = (lane >> 4) << 1;   // K sub-offset 0 or 2
  const int nlan = lane & 15;          // B/C-frag col
  const int mbase = (lane >> 4) << 3;  // C-frag row base 0 or 8

  // GEMM1: Hs(16xH) = tanh(Xs(16x256) @ We1(256x512) + be1)
  for (int nt = 0; nt < 4; ++nt) {
    const int n0 = (wave * 4 + nt) * 16;
    v8f acc = {};
    for (int k0 = 0; k0 < Pn; k0 += 4) {
      v2f a, bb;
      a.x = Xs[mrow * Pn + k0 + koff];
      a.y = Xs[mrow * Pn + k0 + koff + 1];
      bb.x = We1[(size_t)(k0 + koff) * Hn + n0 + nlan];
      bb.y = We1[(size_t)(k0 + koff + 1) * Hn + n0 + nlan];
      acc = wmma_f32(a, bb, acc);
    }
    const int n = n0 + nlan;
    const float bias = be1[n];
#pragma unroll
    for (int r = 0; r < 8; ++r) Hs[(mbase + r) * Hn + n] = tanhf(acc[r] + bias);
  }
  __syncthreads();

  // GEMM2: y tile(16x128) = Hs @ We2(512x128) + be2 ; one 16-col tile per wave
  {
    const int n0 = wave * 16;
    v8f acc = {};
    for (int k0 = 0; k0 < Hn; k0 += 4) {
      v2f a, bb;
      a.x = Hs[mrow * Hn + k0 + koff];
      a.y = Hs[mrow * Hn + k0 + koff + 1];
      bb.x = We2[(size_t)(k0 + koff) * Ln + n0 + nlan];
      bb.y = We2[(size_t)(k0 + koff + 1) * Ln + n0 + nlan];
      acc = wmma_f32(a, bb, acc);
    }
    const int n = n0 + nlan;
    const float bias = be2[n];
#pragma unroll
    for (int r = 0; r < 8; ++r)
      y[((size_t)b * Ln + n) * Tn + t0 + mbase + r] = acc[r] + bias;
  }
}

// ---------------------------------------------------------------------------
// Decode: xr[b,p,t] = (tanh(yin[b,:,t]^T @ Wd1 + bd1) @ Wd2 + bd2)[p]
// grid (B, T/16), block 256.
// ---------------------------------------------------------------------------
__global__ __launch_bounds__(256) void k_decode(
    const float* __restrict__ yin, const float* __restrict__ Wd1,
    const float* __restrict__ bd1, const float* __restrict__ Wd2,
    const float* __restrict__ bd2, float* __restrict__ xr) {
  __shared__ float Ys[16 * Ln];   // 8KB
  __shared__ float Hs[16 * Hn];   // 32KB
  const int b = blockIdx.x;
  const int t0 = blockIdx.y * 16;
  const int tid = threadIdx.x;
  for (int idx = tid; idx < 16 * Ln; idx += 256) {
    int m = idx & 15, l = idx >> 4;
    Ys[m * Ln + l] = yin[((size_t)b * Ln + l) * Tn + t0 + m];
  }
  __syncthreads();
  const int wave = tid >> 5;
  const int lane = tid & 31;
  const int mrow = lane & 15;
  const int koff = (lane >> 4) << 1;
  const int nlan = lane & 15;
  const int mbase = (lane >> 4) << 3;

  // GEMM1: Hs(16x512) = tanh(Ys(16x128) @ Wd1(128x512) + bd1)
  for (int nt = 0; nt < 4; ++nt) {
    const int n0 = (wave * 4 + nt) * 16;
    v8f acc = {};
    for (int k0 = 0; k0 < Ln; k0 += 4) {
      v2f a, bb;
      a.x = Ys[mrow * Ln + k0 + koff];
      a.y = Ys[mrow * Ln + k0 + koff + 1];
      bb.x = Wd1[(size_t)(k0 + koff) * Hn + n0 + nlan];
      bb.y = Wd1[(size_t)(k0 + koff + 1) * Hn + n0 + nlan];
      acc = wmma_f32(a, bb, acc);
    }
    const int n = n0 + nlan;
    const float bias = bd1[n];
#pragma unroll
    for (int r = 0; r < 8; ++r) Hs[(mbase + r) * Hn + n] = tanhf(acc[r] + bias);
  }
  __syncthreads();

  // GEMM2: out(16x256) = Hs @ Wd2(512x256) + bd2 ; 2 tiles per wave
  for (int nt = 0; nt < 2; ++nt) {
    const int n0 = (wave * 2 + nt) * 16;
    v8f acc = {};
    for (int k0 = 0; k0 < Hn; k0 += 4) {
      v2f a, bb;
      a.x = Hs[mrow * Hn + k0 + koff];
      a.y = Hs[mrow * Hn + k0 + koff + 1];
      bb.x = Wd2[(size_t)(k0 + koff) * Pn + n0 + nlan];
      bb.y = Wd2[(size_t)(k0 + koff + 1) * Pn + n0 + nlan];
      acc = wmma_f32(a, bb, acc);
    }
    const int n = n0 + nlan;
    const float bias = bd2[n];
#pragma unroll
    for (int r = 0; r < 8; ++r)
      xr[((size_t)b * Pn + n) * Tn + t0 + mbase + r] = acc[r] + bias;
  }
}

// ---------------------------------------------------------------------------
// Gram: G = ym ym^T, C = yp ym^T (per sample, 128x128 each). K = 255 (masked).
// grid B, block 256; 64 tiles, 8 per wave, computing both G and C.
// ---------------------------------------------------------------------------
__global__ __launch_bounds__(256) void k_gram(const float* __restrict__ y,
                                              float* __restrict__ G,
                                              float* __restrict__ C) {
  const int b = blockIdx.x;
  const float* yb = y + (size_t)b * Ln * Tn;
  float* Gb = G + (size_t)b * Ln * Ln;
  float* Cb = C + (size_t)b * Ln * Ln;
  const int wave = threadIdx.x >> 5;
  const int lane = threadIdx.x & 31;
  const int mrow = lane & 15;
  const int koff = (lane >> 4) << 1;
  const int nlan = lane & 15;
  const int mbase = (lane >> 4) << 3;
  for (int s = 0; s < 8; ++s) {
    const int tile = wave * 8 + s;
    const int i0 = (tile >> 3) * 16;
    const int j0 = (tile & 7) * 16;
    v8f accG = {}, accC = {};
    for (int k0 = 0; k0 < 256; k0 += 4) {
      const int ka = k0 + koff, kb = ka + 1;
      v2f a, a2, bb;
      a.x = (ka < Tn - 1) ? yb[(i0 + mrow) * Tn + ka] : 0.f;       // ym rows
      a.y = (kb < Tn - 1) ? yb[(i0 + mrow) * Tn + kb] : 0.f;
      a2.x = (ka < Tn - 1) ? yb[(i0 + mrow) * Tn + ka + 1] : 0.f;  // yp rows
      a2.y = (kb < Tn - 1) ? yb[(i0 + mrow) * Tn + kb + 1] : 0.f;
      bb.x = (ka < Tn - 1) ? yb[(j0 + nlan) * Tn + ka] : 0.f;      // ym^T cols
      bb.y = (kb < Tn - 1) ? yb[(j0 + nlan) * Tn + kb] : 0.f;
      accG = wmma_f32(a, bb, accG);
      accC = wmma_f32(a2, bb, accC);
    }
#pragma unroll
    for (int r = 0; r < 8; ++r) {
      Gb[(size_t)(i0 + mbase + r) * Ln + j0 + nlan] = accG[r];
      Cb[(size_t)(i0 + mbase + r) * Ln + j0 + nlan] = accC[r];
    }
  }
}

// ---------------------------------------------------------------------------
// In-place Gauss-Jordan inversion of G (SPD, no pivoting). One block / sample.
// Dynamic LDS: 128*128*4 = 64KB.
// ---------------------------------------------------------------------------
__global__ __launch_bounds__(256) void k_gjinv(float* __restrict__ G) {
  extern __shared__ float M[];  // 128x128
  __shared__ float colf[Ln];
  const int b = blockIdx.x;
  float* Gb = G + (size_t)b * Ln * Ln;
  const int tid = threadIdx.x;
  for (int idx = tid; idx < Ln * Ln; idx += 256) M[idx] = Gb[idx];
  __syncthreads();
  for (int k = 0; k < Ln; ++k) {
    const float pinv = 1.0f / M[k * Ln + k];
    if (tid < Ln) colf[tid] = M[tid * Ln + k];  // capture column k
    __syncthreads();
    if (tid < Ln)
      M[k * Ln + tid] = (tid == k) ? pinv : M[k * Ln + tid] * pinv;
    __syncthreads();
    for (int idx = tid; idx < Ln * Ln; idx += 256) {
      const int i = idx >> 7, j = idx & (Ln - 1);
      if (i == k) continue;
      const float f = colf[i];
      M[idx] = (j == k) ? (-f * pinv) : (M[idx] - f * M[k * Ln + j]);
    }
    __syncthreads();
  }
  for (int idx = tid; idx < Ln * Ln; idx += 256) Gb[idx] = M[idx];
}

// ---------------------------------------------------------------------------
// A = C @ Ginv, per-sample 128x128x128. grid B, block 256, 8 tiles/wave.
// ---------------------------------------------------------------------------
__global__ __launch_bounds__(256) void k_amat(const float* __restrict__ C,
                                              const float* __restrict__ Gi,
                                              float* __restrict__ A) {
  const int b = blockIdx.x;
  const float* Cb = C + (size_t)b * Ln * Ln;
  const float* Gb = Gi + (size_t)b * Ln * Ln;
  float* Ab = A + (size_t)b * Ln * Ln;
  const int wave = threadIdx.x >> 5;
  const int lane = threadIdx.x & 31;
  const int mrow = lane & 15;
  const int koff = (lane >> 4) << 1;
  const int nlan = lane & 15;
  const int mbase = (lane >> 4) << 3;
  for (int s = 0; s < 8; ++s) {
    const int tile = wave * 8 + s;
    const int i0 = (tile >> 3) * 16;
    const int j0 = (tile & 7) * 16;
    v8f acc = {};
    for (int k0 = 0; k0 < Ln; k0 += 4) {
      v2f a, bb;
      a.x = Cb[(i0 + mrow) * Ln + k0 + koff];
      a.y = Cb[(i0 + mrow) * Ln + k0 + koff + 1];
      bb.x = Gb[(size_t)(k0 + koff) * Ln + j0 + nlan];
      bb.y = Gb[(size_t)(k0 + koff + 1) * Ln + j0 + nlan];
      acc = wmma_f32(a, bb, acc);
    }
#pragma unroll
    for (int r = 0; r < 8; ++r)
      Ab[(size_t)(i0 + mbase + r) * Ln + j0 + nlan] = acc[r];
  }
}

// ---------------------------------------------------------------------------
// dmd residual: sum over b of || yp - A @ ym ||^2  (atomic accumulate).
// grid B, block 256; wave = i-tile, 16 t-tiles per wave.
// ---------------------------------------------------------------------------
__global__ __launch_bounds__(256) void k_dloss(const float* __restrict__ A,
                                               const float* __restrict__ y,
                                               float* __restrict__ acc_dmd) {
  const int b = blockIdx.x;
  const float* Ab = A + (size_t)b * Ln * Ln;
  const float* yb = y + (size_t)b * Ln * Tn;
  const int wave = threadIdx.x >> 5;
  const int lane = threadIdx.x & 31;
  const int mrow = lane & 15;
  const int koff = (lane >> 4) << 1;
  const int nlan = lane & 15;
  const int mbase = (lane >> 4) << 3;
  float lsum = 0.f;
  const int i0 = wave * 16;
  for (int s = 0; s < 16; ++s) {
    const int tt0 = s * 16;
    v8f acc = {};
    for (int k0 = 0; k0 < Ln; k0 += 4) {
      v2f a, bb;
      a.x = Ab[(i0 + mrow) * Ln + k0 + koff];
      a.y = Ab[(i0 + mrow) * Ln + k0 + koff + 1];
      bb.x = yb[(size_t)(k0 + koff) * Tn + tt0 + nlan];      // ym[k][t]
      bb.y = yb[(size_t)(k0 + koff + 1) * Tn + tt0 + nlan];
      acc = wmma_f32(a, bb, acc);
    }
    const int t = tt0 + nlan;
    if (t < Tn - 1) {
#pragma unroll
      for (int r = 0; r < 8; ++r) {
        const float res = yb[(size_t)(i0 + mbase + r) * Tn + t + 1] - acc[r];
        lsum += res * res;
      }
    }
  }
  atomicAdd(acc_dmd, lsum);
}

// ---------------------------------------------------------------------------
// Rollout: y_pred[:, :, t] = A^t y0. grid B, block 128.
// ---------------------------------------------------------------------------
__global__ __launch_bounds__(128) void k_rollout(const float* __restrict__ A,
                                                 const float* __restrict__ y,
                                                 float* __restrict__ ypred) {
  __shared__ float sv[Ln];
  const int b = blockIdx.x;
  const int i = threadIdx.x;
  const float* Ab = A + (size_t)b * Ln * Ln;
  const float* yb = y + (size_t)b * Ln * Tn;
  float* yp = ypred + (size_t)b * Ln * Tn;
  float cur = yb[(size_t)i * Tn];
  yp[(size_t)i * Tn] = cur;
  const float* Ar = Ab + (size_t)i * Ln;
  for (int t = 1; t < Tn; ++t) {
    sv[i] = cur;
    __syncthreads();
    float a = 0.f;
    for (int j = 0; j < Ln; ++j) a = fmaf(Ar[j], sv[j], a);
    cur = a;
    yp[(size_t)i * Tn + t] = cur;
    __syncthreads();
  }
}

// ---------------------------------------------------------------------------
// Reductions
// ---------------------------------------------------------------------------
__global__ void k_zero(float* d0, float* d1, float* d2, float* d3) {
  if (threadIdx.x == 0) { *d0 = 0.f; *d1 = 0.f; *d2 = 0.f; *d3 = 0.f; }
}

__global__ void k_sqdiff(const float* __restrict__ a,
                         const float* __restrict__ b, int n,
                         float* __restrict__ acc) {
  float s = 0.f;
  for (int i = blockIdx.x * blockDim.x + threadIdx.x; i < n;
       i += gridDim.x * blockDim.x) {
    const float d = a[i] - b[i];
    s += d * d;
  }
  atomicAdd(acc, s);
}

__global__ void k_aeloss(const float* __restrict__ x,
                         const float* __restrict__ xae,
                         float* __restrict__ acc) {
  const int idx = blockIdx.x * blockDim.x + threadIdx.x;  // (b,p) flat
  if (idx < Bn * Pn) {
    const float d = x[(size_t)idx * Tn] - xae[(size_t)idx * Tn];
    atomicAdd(acc, d * d);
  }
}

__global__ void k_final(float* dmd, float* ae, float* pred, float* pdec) {
  if (threadIdx.x == 0) {
    *dmd *= (1.0f / (float)Bn);
    *ae *= (1.0f / (float)(Bn * Pn));
    *pred *= (1.0f / (float)((size_t)Bn * Ln * Tn));
    *pdec *= (1.0f / (float)((size_t)Bn * Pn * Tn));
  }
}

// ---------------------------------------------------------------------------
extern "C" void kernel_launch(void* const* d_in, const int* in_sizes, int n_in,
                              void* d_out, int out_size, void* d_ws,
                              size_t ws_size, hipStream_t stream) {
  (void)in_sizes; (void)n_in; (void)out_size; (void)ws_size;
  const float* x = (const float*)d_in[0];
  const float* We1 = (const float*)d_in[1];
  const float* be1 = (const float*)d_in[2];
  const float* We2 = (const float*)d_in[3];
  const float* be2 = (const float*)d_in[4];
  const float* Wd1 = (const float*)d_in[5];
  const float* bd1 = (const float*)d_in[6];
  const float* Wd2 = (const float*)d_in[7];
  const float* bd2 = (const float*)d_in[8];

  float* out = (float*)d_out;
  // Output layout (flat, return order)
  float* o_xae = out;                                  // B*P*T
  float* o_y = o_xae + (size_t)Bn * Pn * Tn;           // B*L*T
  float* o_dmd = o_y + (size_t)Bn * Ln * Tn;           // 1
  float* o_ae = o_dmd + 1;                             // 1
  float* o_ypred = o_ae + 1;                           // B*L*T
  float* o_pred = o_ypred + (size_t)Bn * Ln * Tn;      // 1
  float* o_ypdec = o_pred + 1;                         // B*P*T
  float* o_pdec = o_ypdec + (size_t)Bn * Pn * Tn;      // 1

  float* ws = (float*)d_ws;
  float* Gm = ws;                                      // B*L*L
  float* Cm = Gm + (size_t)Bn * Ln * Ln;               // B*L*L
  float* Am = Cm + (size_t)Bn * Ln * Ln;               // B*L*L

  k_zero<<<1, 32, 0, stream>>>(o_dmd, o_ae, o_pred, o_pdec);

  dim3 gEnc(Bn, Tn / 16);
  k_encode<<<gEnc, 256, 0, stream>>>(x, We1, be1, We2, be2, o_y);
  k_decode<<<gEnc, 256, 0, stream>>>(o_y, Wd1, bd1, Wd2, bd2, o_xae);

  k_gram<<<Bn, 256, 0, stream>>>(o_y, Gm, Cm);
  k_gjinv<<<Bn, 256, Ln * Ln * sizeof(float), stream>>>(Gm);
  k_amat<<<Bn, 256, 0, stream>>>(Cm, Gm, Am);
  k_dloss<<<Bn, 256, 0, stream>>>(Am, o_y, o_dmd);

  k_rollout<<<Bn, 128, 0, stream>>>(Am, o_y, o_ypred);
  k_decode<<<gEnc, 256, 0, stream>>>(o_ypred, Wd1, bd1, Wd2, bd2, o_ypdec);

  k_aeloss<<<(Bn * Pn + 255) / 256, 256, 0, stream>>>(x, o_xae, o_ae);
  k_sqdiff<<<1024, 256, 0, stream>>>(o_y, o_ypred, Bn * Ln * Tn, o_pred);
  k_sqdiff<<<1024, 256, 0, stream>>>(x, o_ypdec, Bn * Pn * Tn, o_pdec);

  k_final<<<1, 32, 0, stream>>>(o_dmd, o_ae, o_pred, o_pdec);
}